// SymSync_14396730376488
// MI455X (gfx1250) — compile-verified
//
#include <hip/hip_runtime.h>
#include <math.h>

// ---------------------------------------------------------------------------
// Gardner TED + cubic Farrow symbol sync, 2 sps.
//   Phase 1 (WMMA): Farrow coefficients c2,c1,c0 = B_MAT(3x4) x window(4) are a
//     state-independent 4-tap FIR -> batched as 16-sample tiles through
//     V_WMMA_F32_16X16X4_F32 (A = B_MAT padded 16x4, B = tap windows 4x16).
//   Phase 2 (scan): per-channel sequential loop, 1 lane per channel, minimal
//     loop-carried chain: xI = (c2*mu+c1)*mu+c0, TED, PI filter, NCO.
// ---------------------------------------------------------------------------

#define NS   32768
#define K1f  (-0.00246f)
#define K2f  (-8.2e-06f)

typedef __attribute__((ext_vector_type(2))) float v2f;
typedef __attribute__((ext_vector_type(8))) float v8f;

// B_MAT = 0.5 * flip(rows,axis=1) of [[1,-1,-1,1],[-1,3,-1,-1],[0,0,2,0]]
//   row0 (mu^2): [ 0.5, -0.5, -0.5,  0.5]
//   row1 (mu^1): [-0.5, -0.5,  1.5, -0.5]
//   row2 (mu^0): [ 0.0,  1.0,  0.0,  0.0]
__constant__ float BMC[12] = { 0.5f, -0.5f, -0.5f,  0.5f,
                              -0.5f, -0.5f,  1.5f, -0.5f,
                               0.0f,  1.0f,  0.0f,  0.0f };

#if __has_builtin(__builtin_amdgcn_wmma_f32_16x16x4_f32)
#define HAVE_WMMA_F32X4 1
#else
#define HAVE_WMMA_F32X4 0
#endif

// ---------------------------------------------------------------------------
// Phase 1: coefficient precompute.
// ws layout per sample: [c2re, c1re, c0re, c2im, c1im, c0im]  (6 floats)
// D = A(16x4) x B(4x16):
//   A rows M=0..2 = B_MAT rows, rows 3..15 zero.
//     ISA f32 A 16x4 layout: lane<16 -> M=lane, v0=K0, v1=K1;
//                            lane>=16 -> M=lane-16, v0=K2, v1=K3.
//   B col N = taps of sample n0+N: lane<16 -> v0=x[n-3], v1=x[n-2];
//                                  lane>=16 -> v0=x[n-1], v1=x[n].
//   D: lane n (<16) holds coeff row r for sample n0+n in d[r].
// Boundary (taps before x[0]) only exists for the n0==0 tile of each channel,
// and n0 is wave-uniform -> uniform scalar branch, interior path branch-free.
// ---------------------------------------------------------------------------
__global__ void farrow_coeff_wmma(const float* __restrict__ x,
                                  float* __restrict__ ws, int B) {
  const int lane = threadIdx.x & 31;
  const int half = lane >> 4;        // 0: K=0,1  1: K=2,3
  const int M    = lane & 15;
  const int wavesPerBlock = blockDim.x >> 5;
  const int wave   = blockIdx.x * wavesPerBlock + (threadIdx.x >> 5);
  const int nwaves = gridDim.x * wavesPerBlock;
  const int tilesPerCh = NS / 16;
  const int totalTiles = B * tilesPerCh;

  // A operand: constant Farrow matrix (loop-invariant)
  float a0 = 0.f, a1 = 0.f;
  if (M < 3) {
    a0 = BMC[M * 4 + (half ? 2 : 0)];
    a1 = BMC[M * 4 + (half ? 3 : 1)];
  }
#if HAVE_WMMA_F32X4
  v2f A; A.x = a0; A.y = a1;
#endif

  for (int t = wave; t < totalTiles; t += nwaves) {
    const int b  = t / tilesPerCh;
    const int n0 = (t - b * tilesPerCh) * 16;
    const float* xb = x + (size_t)b * NS * 2;
    // taps: sample index s (slot0) and s+1 (slot1); zero-clamp causal start
    const int s = n0 + M - 3 + (half ? 2 : 0);
    float r0, i0, r1, i1;
    if (n0 != 0) {                      // wave-uniform fast path: s >= 13
      const float2 p = *(const float2*)(xb + 2 * s);
      const float2 q = *(const float2*)(xb + 2 * (s + 1));
      r0 = p.x; i0 = p.y; r1 = q.x; i1 = q.y;
    } else {                            // first tile: clamp taps before x[0]
      r0 = 0.f; i0 = 0.f; r1 = 0.f; i1 = 0.f;
      if (s >= 0)     { float2 p = *(const float2*)(xb + 2 * s);       r0 = p.x; i0 = p.y; }
      if (s + 1 >= 0) { float2 q = *(const float2*)(xb + 2 * (s + 1)); r1 = q.x; i1 = q.y; }
    }

#if HAVE_WMMA_F32X4
    v2f Bre; Bre.x = r0; Bre.y = r1;
    v2f Bim; Bim.x = i0; Bim.y = i1;
    v8f z = {};
    v8f dre = __builtin_amdgcn_wmma_f32_16x16x4_f32(false, A, false, Bre,
                                                    (short)0, z, false, false);
    v8f dim_ = __builtin_amdgcn_wmma_f32_16x16x4_f32(false, A, false, Bim,
                                                     (short)0, z, false, false);
    if (half == 0) {
      float* o = ws + ((size_t)b * NS + n0 + M) * 6;
      o[0] = dre[0];  o[1] = dre[1];  o[2] = dre[2];
      o[3] = dim_[0]; o[4] = dim_[1]; o[5] = dim_[2];
    }
#else
    // VALU fallback (compile-safety only)
    if (half == 0) {
      const int n = n0 + M;
      float f[4][2];
      for (int k = 0; k < 4; ++k) {
        int si = n - 3 + k;
        if (si >= 0) { float2 p = *(const float2*)(xb + 2 * si); f[k][0] = p.x; f[k][1] = p.y; }
        else         { f[k][0] = 0.f; f[k][1] = 0.f; }
      }
      float* o = ws + ((size_t)b * NS + n) * 6;
      for (int c = 0; c < 2; ++c) {
        o[0 + 3 * c] = 0.5f * (f[0][c] - f[1][c] - f[2][c] + f[3][c]);
        o[1 + 3 * c] = fmaf(1.5f, f[2][c], -0.5f * (f[0][c] + f[1][c] + f[3][c]));
        o[2 + 3 * c] = f[1][c];
      }
    }
#endif
  }
}

// ---------------------------------------------------------------------------
// Phase 2: the sequential per-channel scan (one lane = one channel).
// Branchless state machine; outputs float4 (y.re, y.im, e, mu) per step.
// ---------------------------------------------------------------------------
__device__ __forceinline__ void scan_step(float c2r, float c1r, float c0r,
                                          float c2i, float c1i, float c0i,
                                          float& mu, float& eta, float& vi,
                                          int& strobe,
                                          float& b0r, float& b0i,
                                          float& b1r, float& b1i,
                                          float4& o4) {
  const float xr = fmaf(fmaf(c2r, mu, c1r), mu, c0r);
  const float xi = fmaf(fmaf(c2i, mu, c1i), mu, c0i);
  const bool isSym = (strobe >> 1) == 1;          // strobe in {2,3}
  const float nanv = __builtin_nanf("");
  const float yr = isSym ? xr : nanv;
  const float yi = isSym ? xi : nanv;
  const float e = (strobe == 2)
      ? fmaf(b0r, b1r - xr, b0i * (b1i - xi)) : 0.f;
  vi = fmaf(K2f, e, vi);
  const float v = fmaf(K1f, e, vi);
  const float W = 0.5f + v;
  // TED buffer update
  const float nb0r = (strobe == 0) ? b0r : xr;
  const float nb0i = (strobe == 0) ? b0i : xi;
  const bool  keep = (strobe == 1) | (strobe == 2);
  const float nb1r = keep ? b0r : ((strobe == 0) ? b1r : 0.f);
  const float nb1i = keep ? b0i : ((strobe == 0) ? b1i : 0.f);
  // NCO
  const float etad = eta - W;
  const bool  fire = etad < 0.f;
  const float muq  = __fdividef(eta, W);
  o4 = make_float4(yr, yi, e, mu);                // emit pre-update mu
  mu     = fire ? muq : mu;
  eta    = fire ? etad + 1.f : etad;
  strobe = (strobe >> 1) + (fire ? 2 : 0);
  b0r = nb0r; b0i = nb0i; b1r = nb1r; b1i = nb1i;
}

__global__ void symsync_scan_coeff(const float* __restrict__ ws,
                                   float* __restrict__ out, int B) {
  const int b = blockIdx.x * blockDim.x + threadIdx.x;
  if (b >= B) return;
  const float* c = ws + (size_t)b * NS * 6;
  float4* o = (float4*)out + (size_t)b * NS;
  float mu = 0.f, eta = 0.f, vi = 0.f;
  int strobe = 0;
  float b0r = 0.f, b0i = 0.f, b1r = 0.f, b1i = 0.f;
#pragma unroll 4
  for (int n = 0; n < NS; ++n) {
    const float* cp = c + (size_t)n * 6;
    __builtin_prefetch(cp + 6 * 64, 0, 0);        // global_prefetch_b8 ahead
    const float2 p0 = *(const float2*)(cp);       // c2r c1r
    const float2 p1 = *(const float2*)(cp + 2);   // c0r c2i
    const float2 p2 = *(const float2*)(cp + 4);   // c1i c0i
    float4 o4;
    scan_step(p0.x, p0.y, p1.x, p1.y, p2.x, p2.y,
              mu, eta, vi, strobe, b0r, b0i, b1r, b1i, o4);
    o[n] = o4;
  }
}

// Fused fallback: computes Farrow coefficients inline (used when ws too small).
__global__ void symsync_scan_fused(const float* __restrict__ x,
                                   float* __restrict__ out, int B) {
  const int b = blockIdx.x * blockDim.x + threadIdx.x;
  if (b >= B) return;
  const float* xb = x + (size_t)b * NS * 2;
  float4* o = (float4*)out + (size_t)b * NS;
  float mu = 0.f, eta = 0.f, vi = 0.f;
  int strobe = 0;
  float b0r = 0.f, b0i = 0.f, b1r = 0.f, b1i = 0.f;
  float f0r = 0.f, f0i = 0.f, f1r = 0.f, f1i = 0.f;
  float f2r = 0.f, f2i = 0.f, f3r = 0.f, f3i = 0.f;
#pragma unroll 4
  for (int n = 0; n < NS; ++n) {
    __builtin_prefetch(xb + 2 * (n + 128), 0, 0);
    const float2 xn = *(const float2*)(xb + 2 * n);
    f0r = f1r; f0i = f1i; f1r = f2r; f1i = f2i;
    f2r = f3r; f2i = f3i; f3r = xn.x; f3i = xn.y;
    const float c2r = 0.5f * ((f0r + f3r) - (f1r + f2r));
    const float c2i = 0.5f * ((f0i + f3i) - (f1i + f2i));
    const float c1r = fmaf(1.5f, f2r, -0.5f * (f0r + f1r + f3r));
    const float c1i = fmaf(1.5f, f2i, -0.5f * (f0i + f1i + f3i));
    const float c0r = f1r, c0i = f1i;
    float4 o4;
    scan_step(c2r, c1r, c0r, c2i, c1i, c0i,
              mu, eta, vi, strobe, b0r, b0i, b1r, b1i, o4);
    o[n] = o4;
  }
}

extern "C" void kernel_launch(void* const* d_in, const int* in_sizes, int n_in,
                              void* d_out, int out_size, void* d_ws, size_t ws_size,
                              hipStream_t stream) {
  const float* x = (const float*)d_in[0];
  const int B = in_sizes[0] / (NS * 2);          // 128 channels
  float* out = (float*)d_out;
  const size_t needWS = (size_t)B * NS * 6 * sizeof(float);
  const int scanBlocks = (B + 31) / 32;
  if (ws_size >= needWS) {
    farrow_coeff_wmma<<<1024, 256, 0, stream>>>(x, (float*)d_ws, B);
    symsync_scan_coeff<<<scanBlocks, 32, 0, stream>>>((const float*)d_ws, out, B);
  } else {
    symsync_scan_fused<<<scanBlocks, 32, 0, stream>>>(x, out, B);
  }
}